// GCN_49074296324300
// MI455X (gfx1250) — compile-verified
//
#include <hip/hip_runtime.h>
#include <hip/hip_bf16.h>

typedef __attribute__((ext_vector_type(2))) float v2f;
typedef __attribute__((ext_vector_type(8))) float v8f;

#define IN_C 64
#define OUT_C 64
#define BN_EPS 1e-5f

// ---------------------------------------------------------------------------
// 1) init: deg = 1 (self loop), stats[0:128] = 0
// ---------------------------------------------------------------------------
__global__ void gcn_init_kernel(unsigned* __restrict__ deg,
                                float* __restrict__ stats, int n) {
    int i = blockIdx.x * blockDim.x + threadIdx.x;
    if (i < n) deg[i] = 1u;          // self-loop contributes 1 to every degree
    if (i < 2 * OUT_C) stats[i] = 0.0f;
}

// ---------------------------------------------------------------------------
// 2) degree on destination
// ---------------------------------------------------------------------------
__global__ void gcn_deg_kernel(const long long* __restrict__ dst,
                               unsigned* __restrict__ deg, int E) {
    int e = blockIdx.x * blockDim.x + threadIdx.x;
    if (e < E) atomicAdd(&deg[(int)dst[e]], 1u);
}

// ---------------------------------------------------------------------------
// 3) deg^-1/2 (deg >= 1 always, self loop)
// ---------------------------------------------------------------------------
__global__ void gcn_dinv_kernel(const unsigned* __restrict__ deg,
                                float* __restrict__ dinv, int n) {
    int i = blockIdx.x * blockDim.x + threadIdx.x;
    if (i < n) dinv[i] = rsqrtf((float)deg[i]);
}

// ---------------------------------------------------------------------------
// 4) h = x @ W^T via V_WMMA_F32_16X16X4_F32 (fp32 WMMA, wave32).
//    One wave computes a 16-node x 64-channel tile.
//    A (16x4 f32): lane l<16 -> M=l, VGPRj holds K=j ; lanes 16-31 -> K=j+2.
//    B (4x16 f32): mirrored -> element (K = j + 2*(lane>>4), N = lane&15).
//    C/D (16x16 f32): VGPRj -> M = j + 8*(lane>>4), N = lane&15.
// ---------------------------------------------------------------------------
__global__ __launch_bounds__(256) void gcn_gemm_wmma_kernel(
        const float* __restrict__ x, const float* __restrict__ W,
        float* __restrict__ h, int n_nodes) {
    // Stage W in LDS as (k-pair, n) float2 tiles: sWp[kp*64+n] = {W[n][2kp], W[n][2kp+1]}
    __shared__ v2f sWp[32 * OUT_C];   // 16 KB
    for (int i = threadIdx.x; i < 32 * OUT_C; i += blockDim.x) {
        int kp = i >> 6;
        int n  = i & 63;
        v2f w;
        w.x = W[n * IN_C + 2 * kp];
        w.y = W[n * IN_C + 2 * kp + 1];
        sWp[i] = w;
    }
    __syncthreads();

    const int lane = threadIdx.x & 31;
    const int wave = threadIdx.x >> 5;
    const int row_base = (blockIdx.x * (blockDim.x >> 5) + wave) * 16;
    if (row_base >= n_nodes) return;   // n_nodes % 16 == 0: tiles are full or absent

    const int m  = lane & 15;   // M-row (for A) / N-col (for B, C/D)
    const int kh = lane >> 4;   // 0 or 1 -> K offset half

    const float* xrow = x + (size_t)(row_base + m) * IN_C;

    v8f acc0 = {}, acc1 = {}, acc2 = {}, acc3 = {};

#pragma unroll
    for (int kt = 0; kt < IN_C / 4; ++kt) {      // K tiles of 4
        const int kb = kt * 4 + 2 * kh;          // even -> 8B aligned
        const v2f a = *(const v2f*)(xrow + kb);  // A fragment (2 VGPRs)
        const int kp = kt * 2 + kh;
        const v2f b0 = sWp[kp * OUT_C + ( 0 + m)];
        const v2f b1 = sWp[kp * OUT_C + (16 + m)];
        const v2f b2 = sWp[kp * OUT_C + (32 + m)];
        const v2f b3 = sWp[kp * OUT_C + (48 + m)];
        acc0 = __builtin_amdgcn_wmma_f32_16x16x4_f32(false, a, false, b0, (short)0, acc0, false, false);
        acc1 = __builtin_amdgcn_wmma_f32_16x16x4_f32(false, a, false, b1, (short)0, acc1, false, false);
        acc2 = __builtin_amdgcn_wmma_f32_16x16x4_f32(false, a, false, b2, (short)0, acc2, false, false);
        acc3 = __builtin_amdgcn_wmma_f32_16x16x4_f32(false, a, false, b3, (short)0, acc3, false, false);
    }

#pragma unroll
    for (int j = 0; j < 8; ++j) {
        const size_t r = (size_t)(row_base + j + 8 * kh) * OUT_C;
        h[r +  0 + m] = acc0[j];
        h[r + 16 + m] = acc1[j];
        h[r + 32 + m] = acc2[j];
        h[r + 48 + m] = acc3[j];
    }
}

// ---------------------------------------------------------------------------
// 5) agg = h * dinv^2  (self-loop term; also initializes the accumulator)
// ---------------------------------------------------------------------------
__global__ void gcn_selfloop_kernel(const float* __restrict__ h,
                                    const float* __restrict__ dinv,
                                    float* __restrict__ agg, int total) {
    int t = blockIdx.x * blockDim.x + threadIdx.x;
    if (t < total) {
        float d = dinv[t >> 6];
        agg[t] = h[t] * d * d;
    }
}

// ---------------------------------------------------------------------------
// 6) edge scatter-add: 16 threads per edge, float4 channels each.
//    agg (25.6 MB) and h (25.6 MB) are L2-resident (192 MB L2) -> atomics
//    stay on-chip.
// ---------------------------------------------------------------------------
__global__ void gcn_scatter_kernel(const long long* __restrict__ src,
                                   const long long* __restrict__ dst,
                                   const float* __restrict__ h,
                                   const float* __restrict__ dinv,
                                   float* __restrict__ agg, int E) {
    int t = blockIdx.x * blockDim.x + threadIdx.x;
    int e = t >> 4;
    if (e >= E) return;
    int c4 = (t & 15) << 2;
    int s = (int)src[e];
    int d = (int)dst[e];
    float norm = dinv[s] * dinv[d];
    const float4 hv = *(const float4*)(h + (size_t)s * OUT_C + c4);
    float* out = agg + (size_t)d * OUT_C + c4;
    atomicAdd(out + 0, hv.x * norm);
    atomicAdd(out + 1, hv.y * norm);
    atomicAdd(out + 2, hv.z * norm);
    atomicAdd(out + 3, hv.w * norm);
}

// ---------------------------------------------------------------------------
// 7) per-channel sum / sumsq over nodes (bias b cancels in BN, so stats are
//    computed directly on agg)
// ---------------------------------------------------------------------------
__global__ __launch_bounds__(256) void gcn_stats_kernel(
        const float* __restrict__ agg, float* __restrict__ stats, int n) {
    const int c  = threadIdx.x & 63;
    const int rl = threadIdx.x >> 6;   // 0..3 row lanes
    float s = 0.0f, s2 = 0.0f;
    for (int r = blockIdx.x * 4 + rl; r < n; r += gridDim.x * 4) {
        float v = agg[(size_t)r * OUT_C + c];
        s  += v;
        s2 += v * v;
    }
    __shared__ float red[2][4][OUT_C];
    red[0][rl][c] = s;
    red[1][rl][c] = s2;
    __syncthreads();
    if (rl == 0) {
        s  = red[0][0][c] + red[0][1][c] + red[0][2][c] + red[0][3][c];
        s2 = red[1][0][c] + red[1][1][c] + red[1][2][c] + red[1][3][c];
        atomicAdd(&stats[c], s);
        atomicAdd(&stats[OUT_C + c], s2);
    }
}

// ---------------------------------------------------------------------------
// 8) BN (biased var) + ReLU.  (agg + b - mean(agg + b)) == agg - mean(agg),
//    so the GCN bias is dropped entirely.
// ---------------------------------------------------------------------------
__global__ void gcn_bn_relu_kernel(const float* __restrict__ agg,
                                   const float* __restrict__ stats,
                                   const float* __restrict__ gamma,
                                   const float* __restrict__ beta,
                                   float* __restrict__ out,
                                   int n, int total) {
    int t = blockIdx.x * blockDim.x + threadIdx.x;
    if (t >= total) return;
    int c = t & 63;
    float inv_n = 1.0f / (float)n;
    float mean  = stats[c] * inv_n;
    float var   = stats[OUT_C + c] * inv_n - mean * mean;
    float y = (agg[t] - mean) * rsqrtf(var + BN_EPS) * gamma[c] + beta[c];
    out[t] = fmaxf(y, 0.0f);
}

// ---------------------------------------------------------------------------
extern "C" void kernel_launch(void* const* d_in, const int* in_sizes, int n_in,
                              void* d_out, int out_size, void* d_ws, size_t ws_size,
                              hipStream_t stream) {
    const float*     x     = (const float*)d_in[0];
    const long long* ei    = (const long long*)d_in[1];   // [2, E] int64
    const float*     W     = (const float*)d_in[2];       // [64, 64]
    // d_in[3] = bias b: cancels under BatchNorm, unused
    const float*     gamma = (const float*)d_in[4];
    const float*     beta  = (const float*)d_in[5];
    float*           out   = (float*)d_out;

    const int N = in_sizes[0] / IN_C;   // 100000
    const int E = in_sizes[1] / 2;      // 1600000
    const long long* src = ei;
    const long long* dst = ei + E;

    // workspace layout
    float*    h     = (float*)d_ws;                       // N*64 f32
    float*    agg   = h + (size_t)N * OUT_C;              // N*64 f32
    unsigned* deg   = (unsigned*)(agg + (size_t)N * OUT_C); // N u32
    float*    dinv  = (float*)(deg + N);                  // N f32
    float*    stats = dinv + N;                           // 128 f32

    const int total = N * OUT_C;

    gcn_init_kernel<<<(N + 255) / 256, 256, 0, stream>>>(deg, stats, N);
    gcn_deg_kernel<<<(E + 255) / 256, 256, 0, stream>>>(dst, deg, E);
    gcn_dinv_kernel<<<(N + 255) / 256, 256, 0, stream>>>(deg, dinv, N);

    // 8 waves/block * 16 rows/wave = 128 rows per block
    gcn_gemm_wmma_kernel<<<(N + 127) / 128, 256, 0, stream>>>(x, W, h, N);

    gcn_selfloop_kernel<<<(total + 255) / 256, 256, 0, stream>>>(h, dinv, agg, total);

    const long long et = (long long)E * 16;
    gcn_scatter_kernel<<<(unsigned)((et + 255) / 256), 256, 0, stream>>>(
        src, dst, h, dinv, agg, E);

    gcn_stats_kernel<<<1024, 256, 0, stream>>>(agg, stats, N);

    gcn_bn_relu_kernel<<<(total + 255) / 256, 256, 0, stream>>>(
        agg, stats, gamma, beta, out, N, total);
}